// GCN_17489106829800
// MI455X (gfx1250) — compile-verified
//
#include <hip/hip_runtime.h>
#include <hip/hip_bf16.h>

typedef __attribute__((ext_vector_type(2))) float v2f;
typedef __attribute__((ext_vector_type(8))) float v8f;

#define D 64
#define WT_STRIDE 68   // transposed-W LDS row stride: conflict-free b64 reads

// ---------------------------------------------------------------------------
// init: deg=1 (self loop), pooled sums/counts = 0
// ---------------------------------------------------------------------------
__global__ void init_kernel(float* __restrict__ deg, float* __restrict__ sums,
                            float* __restrict__ counts, int n) {
    int i = blockIdx.x * blockDim.x + threadIdx.x;
    if (i < n) deg[i] = 1.0f;
    if (i < 128 * 64) sums[i] = 0.0f;
    if (i < 128) counts[i] = 0.0f;
}

// deg[dst] += 1 per edge
__global__ void degree_kernel(const int* __restrict__ dst, float* __restrict__ deg, int E) {
    int e = blockIdx.x * blockDim.x + threadIdx.x;
    if (e < E) atomicAdd(&deg[dst[e]], 1.0f);
}

// deg -> dinv in place
__global__ void dinv_kernel(float* __restrict__ deg, int n) {
    int i = blockIdx.x * blockDim.x + threadIdx.x;
    if (i < n) deg[i] = rsqrtf(deg[i]);
}

// ---------------------------------------------------------------------------
// GEMM: h = X @ W  (f32 WMMA 16x16x4), epilogue: agg = h*dinv^2 + b
// Block: 256 threads = 8 waves; each wave computes a 16x64 tile of h.
// W is staged TRANSPOSED in LDS so each B fragment is one contiguous
// ds_load_b64 into an even-aligned VGPR pair (no shuffle moves, no bank
// conflicts with stride 68).
// ---------------------------------------------------------------------------
__global__ void __launch_bounds__(256)
gemm_kernel(const float* __restrict__ X, const float* __restrict__ W,
            const float* __restrict__ bvec, const float* __restrict__ dinv,
            float* __restrict__ h, float* __restrict__ agg, int n) {
    __shared__ float Wt[D * WT_STRIDE];   // Wt[j][k] = W[k][j]
    // stage: consecutive threads write consecutive LDS addresses (no LDS
    // conflicts); strided global reads of the 16KB W stay hot in L2.
    for (int i = threadIdx.x; i < D * D; i += 256) {
        int j = i >> 6, k = i & 63;
        Wt[j * WT_STRIDE + k] = W[k * D + j];
    }
    __syncthreads();

    const int wave = threadIdx.x >> 5;
    const int lane = threadIdx.x & 31;
    const int half = lane >> 4;     // 0: lanes 0-15, 1: lanes 16-31
    const int lm   = lane & 15;

    const int rowBase = blockIdx.x * 128 + wave * 16;
    int arow = rowBase + lm;                 // A-fragment row for this lane
    if (arow >= n) arow = n - 1;             // clamp (stores guarded below)

    v8f acc[4] = {};                         // 4 col-tiles of 16x16 f32

    for (int k0 = 0; k0 < D; k0 += 4) {
        const int kk = k0 + 2 * half;        // this lane's K pair: kk, kk+1
        // A(16x4): lane l holds A[l%16][kk..kk+1] -> contiguous b64 load
        v2f a = *(const v2f*)(X + (size_t)arow * D + kk);
#pragma unroll
        for (int t = 0; t < 4; ++t) {
            const int n0 = t * 16;
            // B(4x16): lane l holds W[kk][n0+lm], W[kk+1][n0+lm]
            //        = Wt[n0+lm][kk..kk+1] -> single ds_load_b64
            v2f b = *(const v2f*)(&Wt[(n0 + lm) * WT_STRIDE + kk]);
            acc[t] = __builtin_amdgcn_wmma_f32_16x16x4_f32(
                false, a, false, b, (short)0, acc[t], false, false);
        }
    }

    // Epilogue: D-tile vgpr r, lane l -> element [r + 8*(l/16)][l%16]
#pragma unroll
    for (int r = 0; r < 8; ++r) {
        const int orow = rowBase + r + 8 * half;
        if (orow < n) {
            const float di = dinv[orow];
            const float di2 = di * di;
#pragma unroll
            for (int t = 0; t < 4; ++t) {
                const int col = t * 16 + lm;
                const float v = acc[t][r];
                const size_t idx = (size_t)orow * D + col;
                h[idx] = v;
                agg[idx] = v * di2 + bvec[col];
            }
        }
    }
}

// ---------------------------------------------------------------------------
// Edge scatter: one wave32 per edge; 2 floats per lane.
// agg[dst] += h[src] * dinv[src]*dinv[dst]
// e is wave-uniform -> src/dst/dinv loads scalarize.
// ---------------------------------------------------------------------------
__global__ void edge_kernel(const float* __restrict__ h, const int* __restrict__ src,
                            const int* __restrict__ dst, const float* __restrict__ dinv,
                            float* __restrict__ agg, int E) {
    int t = blockIdx.x * blockDim.x + threadIdx.x;
    int e = t >> 5;
    if (e >= E) return;
    int lane = t & 31;
    int s = src[e];
    int d = dst[e];
    float norm = dinv[s] * dinv[d];
    int f = lane * 2;
    const float2 m = *(const float2*)(h + (size_t)s * D + f);
    float* ap = agg + (size_t)d * D + f;
    atomicAdd(ap + 0, m.x * norm);
    atomicAdd(ap + 1, m.y * norm);
}

// ReLU, float4-vectorized: xnext = max(agg, 0)
__global__ void relu_kernel(const float* __restrict__ agg, float* __restrict__ xn, int n4) {
    int i = blockIdx.x * blockDim.x + threadIdx.x;
    if (i < n4) {
        float4 v = ((const float4*)agg)[i];
        v.x = fmaxf(v.x, 0.0f);
        v.y = fmaxf(v.y, 0.0f);
        v.z = fmaxf(v.z, 0.0f);
        v.w = fmaxf(v.w, 0.0f);
        ((float4*)xn)[i] = v;
    }
}

// ---------------------------------------------------------------------------
// Pool: batch is sorted -> run-length accumulate in registers, flush with
// atomics only on graph boundary. 256 threads = 4 sub-chunks x 64 features.
// ---------------------------------------------------------------------------
#define PCHUNK 128
__global__ void pool_kernel(const float* __restrict__ h, const int* __restrict__ batch,
                            float* __restrict__ sums, float* __restrict__ counts, int n) {
    int f = threadIdx.x & 63;
    int sub = threadIdx.x >> 6;
    int start = (blockIdx.x * 4 + sub) * PCHUNK;
    if (start >= n) return;
    int end = start + PCHUNK;
    if (end > n) end = n;
    int cur = batch[start];
    float acc = 0.0f, run = 0.0f;
    for (int node = start; node < end; ++node) {
        int g = batch[node];
        if (g != cur) {
            atomicAdd(&sums[cur * 64 + f], acc);
            if (f == 0) atomicAdd(&counts[cur], run);
            acc = 0.0f; run = 0.0f; cur = g;
        }
        acc += h[(size_t)node * D + f];
        run += 1.0f;
    }
    atomicAdd(&sums[cur * 64 + f], acc);
    if (f == 0) atomicAdd(&counts[cur], run);
}

// out[g] = (sums[g]/max(count,1)) @ lin_w + lin_b   -> [128, 2]
__global__ void final_kernel(const float* __restrict__ sums, const float* __restrict__ counts,
                             const float* __restrict__ lin_w, const float* __restrict__ lin_b,
                             float* __restrict__ out) {
    int g = threadIdx.x;   // 128 graphs
    float inv = 1.0f / fmaxf(counts[g], 1.0f);
    float a0 = 0.0f, a1 = 0.0f;
    for (int f = 0; f < 64; ++f) {
        float p = sums[g * 64 + f] * inv;
        a0 += p * lin_w[f * 2 + 0];
        a1 += p * lin_w[f * 2 + 1];
    }
    out[g * 2 + 0] = a0 + lin_b[0];
    out[g * 2 + 1] = a1 + lin_b[1];
}

// ---------------------------------------------------------------------------
extern "C" void kernel_launch(void* const* d_in, const int* in_sizes, int n_in,
                              void* d_out, int out_size, void* d_ws, size_t ws_size,
                              hipStream_t stream) {
    const float* x     = (const float*)d_in[0];
    const int*   ei    = (const int*)d_in[1];
    const int*   batch = (const int*)d_in[2];
    const float* Wl[3] = { (const float*)d_in[3], (const float*)d_in[5], (const float*)d_in[7] };
    const float* bl[3] = { (const float*)d_in[4], (const float*)d_in[6], (const float*)d_in[8] };
    const float* lin_w = (const float*)d_in[9];
    const float* lin_b = (const float*)d_in[10];
    float* out = (float*)d_out;

    const int n = in_sizes[0] / D;
    const int E = in_sizes[1] / 2;
    const int* src = ei;
    const int* dst = ei + E;

    // workspace layout
    float* ws     = (float*)d_ws;
    size_t nd     = (size_t)n * D;
    float* h      = ws;              // [n,64]
    float* agg    = h + nd;          // [n,64]
    float* xbuf   = agg + nd;        // [n,64]
    float* dinv   = xbuf + nd;       // [n]   (deg, then rsqrt in place)
    float* sums   = dinv + n;        // [128,64]
    float* counts = sums + 128 * 64; // [128]

    const int g256n  = (n + 255) / 256;
    const int g256e  = (E + 255) / 256;
    const int gGemm  = (n + 127) / 128;
    const int gEdge  = (E * 32 + 255) / 256;      // one wave per edge
    const int n4     = n * (D / 4);
    const int gRelu  = (n4 + 255) / 256;
    const int gPool  = (n + 4 * PCHUNK - 1) / (4 * PCHUNK);

    // degree -> dinv
    init_kernel<<<g256n, 256, 0, stream>>>(dinv, sums, counts, n);
    degree_kernel<<<g256e, 256, 0, stream>>>(dst, dinv, E);
    dinv_kernel<<<g256n, 256, 0, stream>>>(dinv, n);

    // 3 GCN layers
    const float* xin = x;
    for (int l = 0; l < 3; ++l) {
        gemm_kernel<<<gGemm, 256, 0, stream>>>(xin, Wl[l], bl[l], dinv, h, agg, n);
        edge_kernel<<<gEdge, 256, 0, stream>>>(h, src, dst, dinv, agg, E);
        relu_kernel<<<gRelu, 256, 0, stream>>>(agg, xbuf, n4);
        xin = xbuf;
    }

    // mean pool + classifier
    pool_kernel<<<gPool, 256, 0, stream>>>(xbuf, batch, sums, counts, n);
    final_kernel<<<1, 128, 0, stream>>>(sums, counts, lin_w, lin_b, out);
}